// NormSelfAttention_3040836845907
// MI455X (gfx1250) — compile-verified
//
#include <hip/hip_runtime.h>
#include <math.h>
#include <stdint.h>

// Problem constants (B, H, N, D) = (4, 16, 2048, 128)
#define B_   4
#define H_   16
#define N_   2048
#define D_   128
#define BH_  (B_ * H_)
#define EPS_ 1e-6f

#define NC_   256   // N-rows per block in kv_accum
#define NT_   32    // rows staged in LDS per tile (both kernels)
#define LDSP  136   // LDS row stride in floats: 2*136 mod 64 == 16 -> halves hit disjoint banks

typedef __attribute__((ext_vector_type(2))) float v2f;
typedef __attribute__((ext_vector_type(4))) float v4f;
typedef __attribute__((ext_vector_type(8))) float v8f;

__device__ __forceinline__ float actf(float x) {
    // elu(x) + 1  ==  x > 0 ? x + 1 : exp(x)
    return x > 0.0f ? x + 1.0f : __expf(x);
}

// CDNA5 async global -> LDS copy, 16B per lane, tracked by ASYNCcnt.
// lds_off: LDS byte offset (low 32 bits of generic shared pointer),
// goff: per-lane byte offset, gbase: uniform 64-bit base (SGPR pair).
__device__ __forceinline__ void async_cp16(uint32_t lds_off, const void* gbase, uint32_t goff) {
    asm volatile("global_load_async_to_lds_b128 %0, %1, %2"
                 :
                 : "v"(lds_off), "v"(goff), "s"(gbase)
                 : "memory");
}
__device__ __forceinline__ void wait_async0() {
    asm volatile("s_wait_asynccnt 0x0" ::: "memory");
}

// ---------------------------------------------------------------- zero KV ws
__global__ __launch_bounds__(256) void zero_kv(float* __restrict__ kv) {
    size_t i = (size_t)blockIdx.x * blockDim.x + threadIdx.x;
    v4f z = (v4f)0.0f;
    ((v4f*)kv)[i] = z;
}

// ------------------------------------------------- KV[b,h] += Kf^T @ Vf chunk
// grid: (BH, N_/NC_), block: 256 (8 wave32). Wave w owns d-rows [16w, 16w+16).
// mask^2 is folded into the K-side fragment, so V stages as a pure async copy.
__global__ __launch_bounds__(256) void kv_accum(const float* __restrict__ K,
                                                const float* __restrict__ V,
                                                const float* __restrict__ mask,
                                                float* __restrict__ KV) {
    __shared__ __align__(16) float Vs[2][NT_ * LDSP];

    const int tid  = threadIdx.x;
    const int lane = tid & 31;
    const int w    = tid >> 5;
    const int half = lane >> 4;   // 0: lanes 0-15, 1: lanes 16-31
    const int l16  = lane & 15;
    const int bh   = blockIdx.x;
    const int b    = bh / H_;
    const size_t base   = (size_t)bh * N_ * D_;
    const int    nchunk = blockIdx.y * NC_;
    const int    dblk   = w * 16;
    const float* Vh = V + base;

    v8f acc[8];
#pragma unroll
    for (int et = 0; et < 8; ++et) acc[et] = (v8f)0.0f;

    auto stage = [&](int buf, int n0base) {
#pragma unroll
        for (int i = tid; i < NT_ * (D_ / 4); i += 256) {
            const int row = i >> 5;
            const int col = (i & 31) << 2;
            const uint32_t lds_off = (uint32_t)(uintptr_t)&Vs[buf][row * LDSP + col];
            const uint32_t goff    = (uint32_t)((((n0base + row) << 7) + col) * 4);
            async_cp16(lds_off, Vh, goff);
        }
    };

    stage(0, nchunk);
    wait_async0();
    __syncthreads();

    for (int nt = 0; nt < NC_; nt += NT_) {
        const int buf = (nt / NT_) & 1;
        if (nt + NT_ < NC_) stage(buf ^ 1, nchunk + nt + NT_);   // prefetch next tile
        const float* vs = Vs[buf];
        const int n0base = nchunk + nt;

#pragma unroll
        for (int k = 0; k < NT_; k += 4) {
            const int nrow = n0base + k + 2 * half;
            // A fragment (16x4 f32): M = d (lane), K = n. Coalesced 64B row segments.
            const float* kp = K + base + (size_t)nrow * D_ + dblk + l16;
            const float  m0 = mask[b * N_ + nrow];
            const float  m1 = mask[b * N_ + nrow + 1];
            v2f a;
            a.x = actf(kp[0])  * (m0 * m0);   // mask applied to both Kf and Vf -> m^2
            a.y = actf(kp[D_]) * (m1 * m1);

            const int vr = (k + 2 * half) * LDSP + l16;
#pragma unroll
            for (int et = 0; et < 8; ++et) {
                v2f bb;
                bb.x = vs[vr + et * 16];          // B row k   (+2 for upper half)
                bb.y = vs[vr + LDSP + et * 16];   // B row k+1 (+2 for upper half)
                acc[et] = __builtin_amdgcn_wmma_f32_16x16x4_f32(
                    false, a, false, bb, (short)0, acc[et], false, false);
            }
        }
        wait_async0();
        __syncthreads();
    }

    // Accumulate partial KV into L2-resident workspace.
    float* kvp = KV + ((size_t)bh << 14);
#pragma unroll
    for (int et = 0; et < 8; ++et) {
#pragma unroll
        for (int r = 0; r < 8; ++r) {
            const int d = dblk + r + half * 8;   // C/D layout: VGPR r -> M=r / M=r+8
            const int e = et * 16 + l16;
            atomicAdd(&kvp[(d << 7) + e], acc[et][r]);
        }
    }
}

// -------------------------------------- out = rmsnorm(Qf @ KV), per 128 rows
// grid: (BH, N_/128), block: 256 (8 wave32). Wave w owns n-rows [16w, 16w+16).
__global__ __launch_bounds__(256) void out_kernel(const float* __restrict__ Q,
                                                  const float* __restrict__ KV,
                                                  float* __restrict__ out) {
    __shared__ __align__(16) float KVs[2][NT_ * LDSP];   // 2 x 17.4 KB, double buffered

    const int tid  = threadIdx.x;
    const int lane = tid & 31;
    const int w    = tid >> 5;
    const int half = lane >> 4;
    const int l16  = lane & 15;
    const int bh   = blockIdx.x;
    const size_t base = (size_t)bh * N_ * D_;
    const int    nblk = blockIdx.y * 128 + w * 16;
    const float* kvp  = KV + ((size_t)bh << 14);

    v8f acc[8];
#pragma unroll
    for (int et = 0; et < 8; ++et) acc[et] = (v8f)0.0f;

    auto stage = [&](int buf, int drow0) {
#pragma unroll
        for (int i = tid; i < NT_ * (D_ / 4); i += 256) {
            const int row = i >> 5;
            const int col = (i & 31) << 2;
            const uint32_t lds_off = (uint32_t)(uintptr_t)&KVs[buf][row * LDSP + col];
            const uint32_t goff    = (uint32_t)((((drow0 + row) << 7) + col) * 4);
            async_cp16(lds_off, kvp, goff);
        }
    };

    stage(0, 0);
    wait_async0();
    __syncthreads();

    for (int dt = 0; dt < D_; dt += NT_) {
        const int buf = (dt / NT_) & 1;
        if (dt + NT_ < D_) stage(buf ^ 1, dt + NT_);   // prefetch next d-tile
        const float* vs = KVs[buf];

#pragma unroll
        for (int d0l = 0; d0l < NT_; d0l += 4) {
            const int d0 = dt + d0l;
            // A fragment: M = n (lane), K = d.
            const float* qp = Q + base + (size_t)(nblk + l16) * D_ + d0 + 2 * half;
            v2f a;
            a.x = actf(qp[0]);
            a.y = actf(qp[1]);

            const int vr = (d0l + 2 * half) * LDSP + l16;
#pragma unroll
            for (int et = 0; et < 8; ++et) {
                v2f bb;
                bb.x = vs[vr + et * 16];
                bb.y = vs[vr + LDSP + et * 16];
                acc[et] = __builtin_amdgcn_wmma_f32_16x16x4_f32(
                    false, a, false, bb, (short)0, acc[et], false, false);
            }
        }
        wait_async0();
        __syncthreads();
    }

    // RMS norm per output row, then coalesced stores.
#pragma unroll
    for (int r = 0; r < 8; ++r) {
        float s = 0.0f;
#pragma unroll
        for (int et = 0; et < 8; ++et) { const float v = acc[et][r]; s += v * v; }
        // Row's 128 values live in one 16-lane half: xor masks 1,2,4,8 stay in-half.
        s += __shfl_xor(s, 1);
        s += __shfl_xor(s, 2);
        s += __shfl_xor(s, 4);
        s += __shfl_xor(s, 8);
        const float scale = rsqrtf(s * (1.0f / 128.0f) + EPS_);
        const int n = nblk + r + half * 8;
        float* op = out + base + (size_t)n * D_ + l16;
#pragma unroll
        for (int et = 0; et < 8; ++et)
            op[et * 16] = acc[et][r] * scale;
    }
}

extern "C" void kernel_launch(void* const* d_in, const int* in_sizes, int n_in,
                              void* d_out, int out_size, void* d_ws, size_t ws_size,
                              hipStream_t stream) {
    const float* Q    = (const float*)d_in[0];
    const float* K    = (const float*)d_in[1];
    const float* V    = (const float*)d_in[2];
    const float* mask = (const float*)d_in[3];
    float* out = (float*)d_out;
    float* KV  = (float*)d_ws;   // BH * 128 * 128 floats = 4 MB scratch

    zero_kv<<<dim3((BH_ * D_ * D_ / 4) / 256), 256, 0, stream>>>(KV);
    kv_accum<<<dim3(BH_, N_ / NC_), 256, 0, stream>>>(K, V, mask, KV);
    out_kernel<<<dim3(BH_, N_ / 128), 256, 0, stream>>>(Q, KV, out);
}